// GCN_BAShapes_58961311039942
// MI455X (gfx1250) — compile-verified
//
#include <hip/hip_runtime.h>
#include <math.h>

typedef float v2f __attribute__((ext_vector_type(2)));
typedef float v8f __attribute__((ext_vector_type(8)));

#define HDIM 128
#define WPB 8           // waves per block in GEMM
#define LDSA_STRIDE 132 // conflict-free A-fragment reads
#define LDSW_STRIDE 136 // conflict-free B-fragment reads

// ---------------------------------------------------------------------------
// GEMM: Hout[N,128] = X[N,(lda)] (first 128 cols) @ W[128,128], via f32 WMMA.
// One wave computes a 16x128 output strip. W cached in LDS, A tile in LDS.
// ---------------------------------------------------------------------------
__launch_bounds__(256)
__global__ void gemm_wmma(const float* __restrict__ X, int lda,
                          const float* __restrict__ W,
                          float* __restrict__ Hout, int nTiles) {
    __shared__ float ldsW[HDIM * LDSW_STRIDE];
    __shared__ float ldsA[WPB][16 * LDSA_STRIDE];

    const int tid  = threadIdx.x;
    const int wave = tid >> 5;
    const int lane = tid & 31;

    // Stage W (16384 floats) into LDS, row stride 136. 64 floats/thread.
    for (int i = 0; i < 16; ++i) {
        int idx = (i * 256 + tid) * 4;       // element index in W, float4 granular
        int row = idx >> 7, col = idx & 127;
        float4 v = *(const float4*)(W + idx);
        *(float4*)(&ldsW[row * LDSW_STRIDE + col]) = v;
    }

    const int tile   = blockIdx.x * WPB + wave;
    const bool active = (tile < nTiles);     // wave-uniform
    if (active) {
        const float* Xrow = X + (size_t)tile * 16 * (size_t)lda;
        for (int r = 0; r < 16; ++r) {
            float4 v = *(const float4*)(Xrow + (size_t)r * lda + lane * 4);
            *(float4*)(&ldsA[wave][r * LDSA_STRIDE + lane * 4]) = v;
        }
    }
    __syncthreads();
    if (!active) return;                     // whole wave exits -> EXEC all-1s for WMMA

    const int m     = lane & 15;
    const int khalf = (lane >> 4) * 2;       // 0 or 2 (A/B K placement per ISA layout)
    const float* aRow = &ldsA[wave][m * LDSA_STRIDE];

    v8f acc[8];
    const v8f vzero = {0.f, 0.f, 0.f, 0.f, 0.f, 0.f, 0.f, 0.f};
#pragma unroll
    for (int nt = 0; nt < 8; ++nt) acc[nt] = vzero;

    for (int k4 = 0; k4 < 32; ++k4) {
        const int kb = k4 * 4 + khalf;
        v2f a;
        a.x = aRow[kb];
        a.y = aRow[kb + 1];
        const float* w0 = &ldsW[kb * LDSW_STRIDE];
        const float* w1 = w0 + LDSW_STRIDE;
#pragma unroll
        for (int nt = 0; nt < 8; ++nt) {
            const int n = nt * 16 + m;
            v2f b;
            b.x = w0[n];
            b.y = w1[n];
            acc[nt] = __builtin_amdgcn_wmma_f32_16x16x4_f32(
                false, a, false, b, (short)0, acc[nt], false, false);
        }
    }

    // C/D layout: VGPR r, lanes 0-15 -> M=r, lanes 16-31 -> M=r+8; N = lane&15.
    float* outBase = Hout + (size_t)tile * 16 * HDIM;
    const int mrow = (lane >> 4) * 8;
#pragma unroll
    for (int nt = 0; nt < 8; ++nt) {
#pragma unroll
        for (int r = 0; r < 8; ++r) {
            outBase[(size_t)(mrow + r) * HDIM + nt * 16 + m] = acc[nt][r];
        }
    }
}

// ---------------------------------------------------------------------------
// Degree / dinv
// ---------------------------------------------------------------------------
__global__ void deg_init(float* __restrict__ deg, int n) {
    int i = blockIdx.x * blockDim.x + threadIdx.x;
    if (i < n) deg[i] = 1.0f;                // self-loop contribution
}

__global__ void deg_count(const int* __restrict__ dst, float* __restrict__ deg, int E) {
    int e = blockIdx.x * blockDim.x + threadIdx.x;
    if (e < E) atomicAdd(&deg[dst[e]], 1.0f);
}

__global__ void make_dinv(float* __restrict__ d, int n) {
    int i = blockIdx.x * blockDim.x + threadIdx.x;
    if (i < n) d[i] = rsqrtf(d[i]);          // deg >= 1 always (self-loops)
}

// ---------------------------------------------------------------------------
// agg init with self-loop: agg[i] = h[i] * dinv[i]^2   (thread per float4)
// ---------------------------------------------------------------------------
__global__ void agg_init(const float* __restrict__ h, const float* __restrict__ dinv,
                         float* __restrict__ agg, int n) {
    int t = blockIdx.x * blockDim.x + threadIdx.x;
    if (t >= n * 32) return;
    int node = t >> 5;
    float w = dinv[node];
    w *= w;
    float4 v = ((const float4*)h)[t];
    v.x *= w; v.y *= w; v.z *= w; v.w *= w;
    ((float4*)agg)[t] = v;
}

// ---------------------------------------------------------------------------
// Edge scatter: one wave32 per edge; coalesced h[src] read, f32 atomics to L2.
// ---------------------------------------------------------------------------
__global__ void edge_agg(const int* __restrict__ src, const int* __restrict__ dst,
                         const float* __restrict__ dinv, const float* __restrict__ h,
                         float* __restrict__ agg, int E) {
    int gid  = blockIdx.x * blockDim.x + threadIdx.x;
    int e    = gid >> 5;
    int lane = gid & 31;
    if (e >= E) return;
    int s = src[e], d = dst[e];
    float w = dinv[s] * dinv[d];
    float4 v = ((const float4*)(h + (size_t)s * HDIM))[lane];
    float* ap = agg + (size_t)d * HDIM + lane * 4;
    atomicAdd(ap + 0, v.x * w);
    atomicAdd(ap + 1, v.y * w);
    atomicAdd(ap + 2, v.z * w);
    atomicAdd(ap + 3, v.w * w);
}

// ---------------------------------------------------------------------------
// relu(agg+b), L2-normalize per node (wave32 per node), write into emb slice.
// ---------------------------------------------------------------------------
__global__ void post_norm(const float* __restrict__ agg, const float* __restrict__ bias,
                          float* __restrict__ embSlice, int n) {
    int gid  = blockIdx.x * blockDim.x + threadIdx.x;
    int node = gid >> 5;
    int lane = gid & 31;
    if (node >= n) return;
    float4 v = ((const float4*)(agg + (size_t)node * HDIM))[lane];
    float4 b = ((const float4*)bias)[lane];
    v.x = fmaxf(v.x + b.x, 0.f);
    v.y = fmaxf(v.y + b.y, 0.f);
    v.z = fmaxf(v.z + b.z, 0.f);
    v.w = fmaxf(v.w + b.w, 0.f);
    float ss = v.x * v.x + v.y * v.y + v.z * v.z + v.w * v.w;
#pragma unroll
    for (int off = 16; off > 0; off >>= 1)
        ss += __shfl_xor(ss, off, 32);
    float scale = 1.0f / fmaxf(sqrtf(ss), 1e-12f);
    float4 o;
    o.x = v.x * scale; o.y = v.y * scale; o.z = v.z * scale; o.w = v.w * scale;
    *(float4*)(embSlice + (size_t)node * 384 + lane * 4) = o;
}

// ---------------------------------------------------------------------------
// Classifier: logits = emb @ Wlin + blin, then log_softmax. Thread per node.
// ---------------------------------------------------------------------------
__launch_bounds__(256)
__global__ void classifier(const float* __restrict__ emb, const float* __restrict__ Wlin,
                           const float* __restrict__ blin, float* __restrict__ out, int n) {
    __shared__ float lw[384 * 8];
    __shared__ float lb[8];
    for (int i = threadIdx.x; i < 384 * 8; i += 256) lw[i] = Wlin[i];
    if (threadIdx.x < 8) lb[threadIdx.x] = blin[threadIdx.x];
    __syncthreads();

    int node = blockIdx.x * blockDim.x + threadIdx.x;
    if (node >= n) return;

    float acc[8];
#pragma unroll
    for (int c = 0; c < 8; ++c) acc[c] = lb[c];
    const float* er = emb + (size_t)node * 384;
    for (int k = 0; k < 384; ++k) {
        float e = er[k];
#pragma unroll
        for (int c = 0; c < 8; ++c) acc[c] = fmaf(e, lw[k * 8 + c], acc[c]);
    }
    float m = acc[0];
#pragma unroll
    for (int c = 1; c < 8; ++c) m = fmaxf(m, acc[c]);
    float s = 0.f;
#pragma unroll
    for (int c = 0; c < 8; ++c) s += expf(acc[c] - m);
    float lse = m + logf(s);
#pragma unroll
    for (int c = 0; c < 8; ++c) out[(size_t)node * 8 + c] = acc[c] - lse;
}

// ---------------------------------------------------------------------------
extern "C" void kernel_launch(void* const* d_in, const int* in_sizes, int n_in,
                              void* d_out, int out_size, void* d_ws, size_t ws_size,
                              hipStream_t stream) {
    const float* x    = (const float*)d_in[0];
    const int*   ei   = (const int*)d_in[1];
    const float* W1   = (const float*)d_in[2];
    const float* b1   = (const float*)d_in[3];
    const float* W2   = (const float*)d_in[4];
    const float* b2   = (const float*)d_in[5];
    const float* W3   = (const float*)d_in[6];
    const float* b3   = (const float*)d_in[7];
    const float* Wlin = (const float*)d_in[8];
    const float* blin = (const float*)d_in[9];

    const int n = in_sizes[0] / HDIM;        // 100000
    const int E = in_sizes[1] / 2;           // 1600000
    const int* src = ei;
    const int* dst = ei + E;

    // Workspace layout (floats): dinv | h | agg | emb
    float* ws   = (float*)d_ws;
    float* dinv = ws;
    float* h    = ws + (size_t)n;                 // 16B-aligned (n multiple of 4)
    float* agg  = h  + (size_t)n * HDIM;
    float* emb  = agg + (size_t)n * HDIM;

    const int T = 256;
    // --- degree / dinv ---
    deg_init<<<(n + T - 1) / T, T, 0, stream>>>(dinv, n);
    deg_count<<<(E + T - 1) / T, T, 0, stream>>>(dst, dinv, E);
    make_dinv<<<(n + T - 1) / T, T, 0, stream>>>(dinv, n);

    const int nTiles = (n + 15) / 16;             // n divisible by 16
    const float* Ws[3] = {W1, W2, W3};
    const float* bs[3] = {b1, b2, b3};
    const float* Xin = x;
    int lda = HDIM;

    for (int l = 0; l < 3; ++l) {
        gemm_wmma<<<(nTiles + WPB - 1) / WPB, 256, 0, stream>>>(Xin, lda, Ws[l], h, nTiles);
        agg_init<<<((size_t)n * 32 + T - 1) / T, T, 0, stream>>>(h, dinv, agg, n);
        edge_agg<<<(unsigned)(((size_t)E * 32 + T - 1) / T), T, 0, stream>>>(src, dst, dinv, h, agg, E);
        post_norm<<<((size_t)n * 32 + T - 1) / T, T, 0, stream>>>(agg, bs[l], emb + l * HDIM, n);
        Xin = emb + l * HDIM;
        lda = 384;
    }

    classifier<<<(n + T - 1) / T, T, 0, stream>>>(emb, Wlin, blin, (float*)d_out, n);
}